// NeuralHMMClassifier_27152783245620
// MI455X (gfx1250) — compile-verified
//
#include <hip/hip_runtime.h>
#include <hip/hip_bf16.h>
#include <math.h>

#define K_STATES 128
#define VOCAB    50257
#define D_EMB    128
#define H_DIM    128
#define B_BATCH  256
#define T_LEN    1024

typedef __attribute__((ext_vector_type(2))) float v2f;
typedef __attribute__((ext_vector_type(8))) float v8f;

__device__ __forceinline__ v8f wmma_f32_4(v2f a, v2f b, v8f c) {
  // D = A(16x4 f32) * B(4x16 f32) + C(16x16 f32), full f32 precision
  return __builtin_amdgcn_wmma_f32_16x16x4_f32(false, a, false, b, (short)0, c,
                                               false, false);
}

// ---------------------------------------------------------------------------
// 1) trans_logits[row] = dot(trans_w[row,:], trans_query) + trans_b[row]
// ---------------------------------------------------------------------------
__global__ void k_trans_logits(const float* __restrict__ trans_w,
                               const float* __restrict__ trans_b,
                               const float* __restrict__ trans_q,
                               float* __restrict__ tl) {
  __shared__ float q[H_DIM];
  int tid = threadIdx.x;
  if (tid < H_DIM) q[tid] = trans_q[tid];
  __syncthreads();
  int row = blockIdx.x * blockDim.x + tid;  // 0..16383
  if (row < K_STATES * K_STATES) {
    const float* w = trans_w + (size_t)row * H_DIM;
    float s = trans_b[row];
#pragma unroll 8
    for (int j = 0; j < H_DIM; ++j) s += w[j] * q[j];
    tl[row] = s;
  }
}

// ---------------------------------------------------------------------------
// 2) A[i,j] = softmax_j(trans_logits[i,:])  (row-stochastic transition probs)
// ---------------------------------------------------------------------------
__global__ void k_rowsoftmax_A(const float* __restrict__ tl,
                               float* __restrict__ A) {
  __shared__ float red[K_STATES];
  int i = blockIdx.x, j = threadIdx.x;
  float x = tl[i * K_STATES + j];
  red[j] = x;
  __syncthreads();
  for (int off = 64; off >= 1; off >>= 1) {
    if (j < off) red[j] = fmaxf(red[j], red[j + off]);
    __syncthreads();
  }
  float mx = red[0];
  __syncthreads();
  float e = expf(x - mx);
  red[j] = e;
  __syncthreads();
  for (int off = 64; off >= 1; off >>= 1) {
    if (j < off) red[j] += red[j + off];
    __syncthreads();
  }
  A[i * K_STATES + j] = e / red[0];
}

// ---------------------------------------------------------------------------
// 3) log_pi = log_softmax(init_logits)
// ---------------------------------------------------------------------------
__global__ void k_logpi(const float* __restrict__ il, float* __restrict__ lp) {
  __shared__ float red[K_STATES];
  int j = threadIdx.x;
  float x = il[j];
  red[j] = x;
  __syncthreads();
  for (int off = 64; off >= 1; off >>= 1) {
    if (j < off) red[j] = fmaxf(red[j], red[j + off]);
    __syncthreads();
  }
  float mx = red[0];
  __syncthreads();
  red[j] = expf(x - mx);
  __syncthreads();
  for (int off = 64; off >= 1; off >>= 1) {
    if (j < off) red[j] += red[j + off];
    __syncthreads();
  }
  lp[j] = x - mx - logf(red[0]);
}

// ---------------------------------------------------------------------------
// 4) Emission scores S[v,k] = dot(word_embed[v,:], tag_embed[k,:]) + bias[v]
//    stored transposed (V x K). LDS-staged: 64 word rows (32KB) + full tag
//    matrix (64KB) per block; branch-free WMMA loop (v clamped, store-side
//    guard only). 8 waves x 4 M-tiles x 32 depth-4 f32 WMMAs per block.
// ---------------------------------------------------------------------------
__global__ void k_emis_scores(const float* __restrict__ Wv,   // (V, D)
                              const float* __restrict__ Tg,   // (K, D)
                              const float* __restrict__ bias, // (V)
                              float* __restrict__ S) {        // (V, K)
  extern __shared__ float esm[];
  float* wtile = esm;                 // 64 * 128
  float* ttile = esm + 64 * D_EMB;    // 128 * 128

  int tid = threadIdx.x;
  int vt = blockIdx.x;                // 64-row super-tile index
  // stage tag matrix (coalesced; L2-resident source)
  for (int idx = tid; idx < K_STATES * D_EMB; idx += 256) ttile[idx] = Tg[idx];
  // stage 64 word rows, clamped at the vocab edge (garbage rows never stored)
  for (int idx = tid; idx < 64 * D_EMB; idx += 256) {
    int vv = vt * 64 + (idx >> 7);
    if (vv >= VOCAB) vv = VOCAB - 1;
    wtile[idx] = Wv[(size_t)vv * D_EMB + (idx & 127)];
  }
  __syncthreads();

  int w = tid >> 5;        // wave 0..7 -> tag column tile
  int L = tid & 31;
  int half = L >> 4;
  int mrow = L & 15;
  int n = w * 16 + mrow;   // tag index (B fragment col == C col)
  const float* trow = ttile + n * D_EMB;

#pragma unroll
  for (int mt = 0; mt < 4; ++mt) {
    const float* wrow = wtile + (mt * 16 + mrow) * D_EMB;
    v8f c = {0.f, 0.f, 0.f, 0.f, 0.f, 0.f, 0.f, 0.f};
#pragma unroll
    for (int ch = 0; ch < 32; ++ch) {
      int kb = ch * 4 + half * 2;
      v2f a = *(const v2f*)(wrow + kb);
      v2f b = *(const v2f*)(trow + kb);  // B[d][n] = Tg[n][d], contiguous
      c = wmma_f32_4(a, b, c);
    }
#pragma unroll
    for (int r = 0; r < 8; ++r) {
      int vm = vt * 64 + mt * 16 + r + half * 8;
      if (vm < VOCAB) S[(size_t)vm * K_STATES + n] = c[r] + bias[vm];
    }
  }
}

// ---------------------------------------------------------------------------
// 5a) Per-column (per tag) partial max / sum-exp over a V chunk
// ---------------------------------------------------------------------------
__global__ void k_col_reduce(const float* __restrict__ S,
                             float* __restrict__ pMax,
                             float* __restrict__ pSum) {
  int k = threadIdx.x;
  int c = blockIdx.x;
  const int CH = (VOCAB + 63) / 64;
  int v0 = c * CH;
  int v1 = min(VOCAB, v0 + CH);
  float mx = -INFINITY;
  for (int v = v0; v < v1; ++v) mx = fmaxf(mx, S[(size_t)v * K_STATES + k]);
  float s = 0.f;
  for (int v = v0; v < v1; ++v) s += expf(S[(size_t)v * K_STATES + k] - mx);
  pMax[c * K_STATES + k] = mx;
  pSum[c * K_STATES + k] = s;
}

// 5b) Combine partials -> lse[k] = logsumexp_v S[v,k]
__global__ void k_col_combine(const float* __restrict__ pMax,
                              const float* __restrict__ pSum,
                              float* __restrict__ lse) {
  int k = threadIdx.x;
  float M = -INFINITY;
  for (int c = 0; c < 64; ++c) M = fmaxf(M, pMax[c * K_STATES + k]);
  float s = 0.f;
  for (int c = 0; c < 64; ++c) {
    float pm = pMax[c * K_STATES + k];
    if (pm > -INFINITY) s += pSum[c * K_STATES + k] * expf(pm - M);
  }
  lse[k] = M + logf(s);
}

// ---------------------------------------------------------------------------
// 6) Persistent scaled-forward scan. 16 WGs x 16 batch rows; A^T + alpha + p
//    resident in LDS; wave w owns output columns [16w,16w+16); 32 depth-4
//    f32 WMMAs per wave per time step. log_B gather folded in: logb = S - lse.
// ---------------------------------------------------------------------------
__global__ void k_forward_scan(const int* __restrict__ emis,    // (B, T)
                               const float* __restrict__ A,     // (K, K)
                               const float* __restrict__ logpi, // (K)
                               const float* __restrict__ S,     // (V, K)
                               const float* __restrict__ lse,   // (K)
                               float* __restrict__ alphaOut) {  // (B, K)
  extern __shared__ float smem[];
  float* At    = smem;                          // 128*128 (transposed A)
  float* alpha = At + K_STATES * K_STATES;      // 16*128
  float* pbuf  = alpha + 16 * K_STATES;         // 16*128
  float* cmax  = pbuf + 16 * K_STATES;          // 16
  int*   tok   = (int*)(cmax + 16);             // 16

  int tid  = threadIdx.x;
  int w    = tid >> 5;
  int L    = tid & 31;
  int half = L >> 4;
  int mrow = L & 15;
  int col  = w * 16 + mrow;
  int b0   = blockIdx.x * 16;

  // A^T into LDS: At[j*128+i] = A[i*128+j] (coalesced global reads)
  for (int idx = tid; idx < K_STATES * K_STATES; idx += 256) {
    int i = idx >> 7, j = idx & 127;
    At[j * K_STATES + i] = A[idx];
  }
  float lse_col = lse[col];

  // alpha_0[b,k] = log_pi[k] + S[e[b,0],k] - lse[k]
  for (int idx = tid; idx < 16 * K_STATES; idx += 256) {
    int m = idx >> 7, k = idx & 127;
    int v = emis[(size_t)(b0 + m) * T_LEN];
    alpha[idx] = logpi[k] + S[(size_t)v * K_STATES + k] - lse[k];
  }

  for (int t = 1; t < T_LEN; ++t) {
    __syncthreads();
    {  // per-row max over 128 cols (16 lanes per row, shfl reduce)
      int row = tid >> 4, ls = tid & 15;
      float mx = -INFINITY;
#pragma unroll
      for (int j = 0; j < 8; ++j)
        mx = fmaxf(mx, alpha[row * K_STATES + ls + j * 16]);
#pragma unroll
      for (int off = 8; off >= 1; off >>= 1)
        mx = fmaxf(mx, __shfl_xor(mx, off, 32));
      if (ls == 0) cmax[row] = mx;
    }
    if (tid < 16) tok[tid] = emis[(size_t)(b0 + tid) * T_LEN + t];
    __syncthreads();
    for (int idx = tid; idx < 16 * K_STATES; idx += 256) {
      int m = idx >> 7;
      pbuf[idx] = __expf(alpha[idx] - cmax[m]);  // p in [0,1]
    }
    __syncthreads();

    v8f c = {0.f, 0.f, 0.f, 0.f, 0.f, 0.f, 0.f, 0.f};
    const float* prow = pbuf + mrow * K_STATES;  // A-frag row m = lane&15
    const float* arow = At + col * K_STATES;     // B-frag: At[col][d] contiguous
#pragma unroll
    for (int ch = 0; ch < 32; ++ch) {
      int kb = ch * 4 + half * 2;
      v2f a = *(const v2f*)(prow + kb);
      v2f b = *(const v2f*)(arow + kb);
      c = wmma_f32_4(a, b, c);
    }
    // alpha_new = log(q) + cmax + (S[tok,col] - lse[col]) straight from fragment
#pragma unroll
    for (int r = 0; r < 8; ++r) {
      int m = r + half * 8;
      int v = tok[m];
      float na = __logf(c[r]) + cmax[m] + (S[(size_t)v * K_STATES + col] - lse_col);
      alpha[m * K_STATES + col] = na;
    }
  }
  __syncthreads();
  for (int idx = tid; idx < 16 * K_STATES; idx += 256) {
    int m = idx >> 7, k = idx & 127;
    alphaOut[(size_t)(b0 + m) * K_STATES + k] = alpha[idx];
  }
}

// ---------------------------------------------------------------------------
// 7) loss = -mean_b logsumexp_k alpha_T[b,k]
// ---------------------------------------------------------------------------
__global__ void k_finalize(const float* __restrict__ alphaOut,
                           float* __restrict__ out) {
  __shared__ float red[B_BATCH];
  int b = threadIdx.x;
  const float* a = alphaOut + (size_t)b * K_STATES;
  float mx = -INFINITY;
  for (int k = 0; k < K_STATES; ++k) mx = fmaxf(mx, a[k]);
  float s = 0.f;
  for (int k = 0; k < K_STATES; ++k) s += expf(a[k] - mx);
  red[b] = mx + logf(s);
  __syncthreads();
  for (int off = 128; off >= 1; off >>= 1) {
    if (b < off) red[b] += red[b + off];
    __syncthreads();
  }
  if (b == 0) out[0] = -red[0] / (float)B_BATCH;
}

// ---------------------------------------------------------------------------
extern "C" void kernel_launch(void* const* d_in, const int* in_sizes, int n_in,
                              void* d_out, int out_size, void* d_ws,
                              size_t ws_size, hipStream_t stream) {
  const int*   emis   = (const int*)d_in[0];
  const float* initlg = (const float*)d_in[1];
  const float* tagw   = (const float*)d_in[2];
  const float* wordw  = (const float*)d_in[3];
  const float* wbias  = (const float*)d_in[4];
  const float* transw = (const float*)d_in[5];
  const float* transb = (const float*)d_in[6];
  const float* transq = (const float*)d_in[7];
  float* out = (float*)d_out;
  (void)in_sizes; (void)n_in; (void)out_size; (void)ws_size;

  char* ws = (char*)d_ws;
  size_t off = 0;
  auto alloc = [&](size_t bytes) {
    void* p = ws + off;
    off += (bytes + 255) & ~(size_t)255;
    return p;
  };
  float* S     = (float*)alloc((size_t)VOCAB * K_STATES * sizeof(float)); // 25.7MB
  float* lse   = (float*)alloc(K_STATES * sizeof(float));
  float* logpi = (float*)alloc(K_STATES * sizeof(float));
  float* Amat  = (float*)alloc(K_STATES * K_STATES * sizeof(float));
  float* tl    = (float*)alloc(K_STATES * K_STATES * sizeof(float));
  float* pMax  = (float*)alloc(64 * K_STATES * sizeof(float));
  float* pSum  = (float*)alloc(64 * K_STATES * sizeof(float));
  float* aOut  = (float*)alloc((size_t)B_BATCH * K_STATES * sizeof(float));

  k_trans_logits<<<(K_STATES * K_STATES) / 256, 256, 0, stream>>>(transw, transb,
                                                                  transq, tl);
  k_rowsoftmax_A<<<K_STATES, K_STATES, 0, stream>>>(tl, Amat);
  k_logpi<<<1, K_STATES, 0, stream>>>(initlg, logpi);

  size_t esmem = (size_t)(64 * D_EMB + K_STATES * D_EMB) * sizeof(float); // 96KB
  k_emis_scores<<<(VOCAB + 63) / 64, 256, esmem, stream>>>(wordw, tagw, wbias, S);
  k_col_reduce<<<64, K_STATES, 0, stream>>>(S, pMax, pSum);
  k_col_combine<<<1, K_STATES, 0, stream>>>(pMax, pSum, lse);

  size_t smem = (size_t)(K_STATES * K_STATES + 2 * 16 * K_STATES + 32) * sizeof(float);
  k_forward_scan<<<B_BATCH / 16, 256, smem, stream>>>(emis, Amat, logpi, S, lse,
                                                      aOut);
  k_finalize<<<1, B_BATCH, 0, stream>>>(aOut, out);
}